// LNCCLoss_40072044871700
// MI455X (gfx1250) — compile-verified
//
#include <hip/hip_runtime.h>
#include <hip/hip_bf16.h>

// ---------------------------------------------------------------------------
// Multi-scale LNCC loss, separable dilated box sums. All shape parameters are
// compile-time template arguments so VMEM uses immediate offsets and the
// dominant 64^3 scale is fully guard-free.
//   passW : (B,D,H,W)->(B,D,H,O)   1-D dilated window sum along W (VALU)
//   passH : (B,D,H,O)->(B,D,O,O)   banded GEMM via V_WMMA_F32_16X16X4_F32
//   passD : (B,D,O,O)->(B,O,O,O)
//   lncc  : pointwise + deterministic 2-stage reduction
// ---------------------------------------------------------------------------

typedef float v2f __attribute__((ext_vector_type(2)));
typedef float v8f __attribute__((ext_vector_type(8)));

#define NDIM   144
#define NB     4
#define NLINES (NB * NDIM * NDIM)       // 82944 (b,d,h) lines
#define NSLABS (NB * NDIM)              // 576 (b,d) slabs
#define SUMS_STRIDE (4 * 64 * 64 * 64)  // 1048576, max per-field output elems
#define EPS_LNCC 1e-5f

// ---------------- pass W: dilated 1-D window sum along W -------------------
// FIELD: 0=i 1=t 2=i*i 3=t*t 4=i*t
template <int FIELD, int K, int ST, int O>
__global__ __launch_bounds__(256)
void lncc_pass_w(const float* __restrict__ A, const float* __restrict__ T,
                 float* __restrict__ out)
{
    constexpr int TOTAL = NLINES * O;
    int tid = blockIdx.x * 256 + threadIdx.x;
    if (tid >= TOTAL) return;
    int line = tid / O;                 // constant divisor -> mul/shift
    int ow   = tid - line * O;
    int base = line * NDIM + ST * ow;

    float s = 0.0f;
#pragma unroll
    for (int j = 0; j < K; ++j) {
        if constexpr (FIELD == 0)      s += A[base + 2 * j];
        else if constexpr (FIELD == 1) s += T[base + 2 * j];
        else if constexpr (FIELD == 2) { float v = A[base + 2 * j]; s = fmaf(v, v, s); }
        else if constexpr (FIELD == 3) { float v = T[base + 2 * j]; s = fmaf(v, v, s); }
        else                           s = fmaf(A[base + 2 * j], T[base + 2 * j], s);
    }
    out[tid] = s;   // layout line*O+ow == tid
}

// ---------------- pass H: banded GEMM with f32 WMMA ------------------------
// tmp2[oh,ow] = sum_h S[oh,h]*tmp1[h,ow], S[oh,h]=1 iff h = ST*oh + 2*j, j<K.
// OOB columns are address-clamped, OOB rows get garbage selection bits: both
// only reach D elements that are never stored. For O%16==0 no guards at all.
template <int K, int ST, int O>
__global__ __launch_bounds__(32)
void lncc_pass_h_wmma(const float* __restrict__ tmp1, float* __restrict__ tmp2)
{
    constexpr int  TILES = (O + 15) / 16;
    constexpr bool FULL  = (O % 16 == 0);
    constexpr unsigned SPAN = (unsigned)(2 * K);    // taps: even r in [0, 2K)

    const int slab = blockIdx.x;                    // b*144 + d
    const int oh0  = ((int)blockIdx.y / TILES) * 16;
    const int ow0  = ((int)blockIdx.y % TILES) * 16;
    const int lane = threadIdx.x;                   // full wave, EXEC all 1s
    const int half = lane >> 4;
    const int l15  = lane & 15;

    const float* __restrict__ src = tmp1 + (size_t)slab * NDIM * O;
    float*       __restrict__ dst = tmp2 + (size_t)slab * O * O;

    const int ow  = ow0 + l15;
    const int owc = FULL ? ow : (ow < O ? ow : O - 1);
    const int kq0 = half ? 2 : 0;                   // K slot of VGPR0 (ISA layout)
    const int kq1 = half ? 3 : 1;                   // K slot of VGPR1
    const int stoh = ST * (oh0 + l15);              // A row for this lane

    const float* p0 = src + kq0 * O + owc;
    const float* p1 = src + kq1 * O + owc;

    // software pipeline: K-step 0 loaded up front; all offsets are immediates
    v2f bcur;
    bcur.x = p0[0];
    bcur.y = p1[0];

    v8f c = {};
#pragma unroll
    for (int kk = 0; kk < 36; ++kk) {               // K = 144 as 36 x (K=4)
        const int hbase = 4 * kk;
        const int r0 = hbase + kq0 - stoh;
        const int r1 = hbase + kq1 - stoh;
        v2f a;
        a.x = ((unsigned)r0 < SPAN && (r0 & 1) == 0) ? 1.0f : 0.0f;
        a.y = ((unsigned)r1 < SPAN && (r1 & 1) == 0) ? 1.0f : 0.0f;

        const v2f b = bcur;
        const int nxt = (kk < 35 ? kk + 1 : 35) * 4 * O;    // compile-time imm
        bcur.x = p0[nxt];
        bcur.y = p1[nxt];

        c = __builtin_amdgcn_wmma_f32_16x16x4_f32(false, a, false, b,
                                                  (short)0, c, false, false);
    }

    // C/D layout: VGPR r -> M=r (lanes 0-15) / M=r+8 (lanes 16-31), N = lane&15
    float* p = dst + (size_t)(oh0 + (half ? 8 : 0)) * O + ow;
    if constexpr (FULL) {
#pragma unroll
        for (int r = 0; r < 8; ++r) p[r * O] = c[r];        // imm offsets
    } else {
        if (ow < O) {
#pragma unroll
            for (int r = 0; r < 8; ++r) {
                int oh = oh0 + r + (half ? 8 : 0);
                if (oh < O) p[r * O] = c[r];
            }
        }
    }
}

// ---------------- pass D: dilated 1-D window sum along D -------------------
template <int K, int ST, int O>
__global__ __launch_bounds__(256)
void lncc_pass_d(const float* __restrict__ tmp2, float* __restrict__ sumf)
{
    constexpr int TOTAL = NB * O * O * O;
    constexpr int PLANE = O * O;
    int tid = blockIdx.x * 256 + threadIdx.x;
    if (tid >= TOTAL) return;
    int ow = tid % O;
    int t  = tid / O;
    int oh = t % O;  t /= O;
    int od = t % O;
    int bb = t / O;

    const float* src = tmp2 + ((size_t)(bb * NDIM + ST * od) * O + oh) * O + ow;
    float s = 0.0f;
#pragma unroll
    for (int j = 0; j < K; ++j) s += src[2 * j * PLANE];    // imm offsets
    sumf[tid] = s;   // ((bb*O+od)*O+oh)*O+ow == tid
}

// ---------------- lncc pointwise + per-block partial sums ------------------
template <int K, int O>
__global__ __launch_bounds__(256)
void lncc_partials(const float* __restrict__ sums, float* __restrict__ partials)
{
    constexpr int   E     = NB * O * O * O;
    constexpr float NUMEL = (float)K * (float)K * (float)K;
    __shared__ float sm[256];
    int tid = blockIdx.x * 256 + threadIdx.x;
    float l = 0.0f;
    if (tid < E) {
        float si  = sums[tid];
        float stt = sums[SUMS_STRIDE + tid];
        float si2 = sums[2 * SUMS_STRIDE + tid];
        float st2 = sums[3 * SUMS_STRIDE + tid];
        float sit = sums[4 * SUMS_STRIDE + tid];
        constexpr float inv = 1.0f / NUMEL;
        float cross = sit - si * stt * inv;
        float iv    = si2 - si * si  * inv;
        float tv    = st2 - stt * stt * inv;
        l = (cross * cross) / (iv * tv + EPS_LNCC);
    }
    sm[threadIdx.x] = l;
    __syncthreads();
    for (int o = 128; o > 0; o >>= 1) {
        if (threadIdx.x < o) sm[threadIdx.x] += sm[threadIdx.x + o];
        __syncthreads();
    }
    if (threadIdx.x == 0) partials[blockIdx.x] = sm[0];
}

// ---------------- deterministic single-block reduce ------------------------
__global__ __launch_bounds__(256)
void lncc_reduce(const float* __restrict__ partials, int n, float* __restrict__ accOut)
{
    __shared__ float sm[256];
    float s = 0.0f;
    for (int i = threadIdx.x; i < n; i += 256) s += partials[i];   // fixed order
    sm[threadIdx.x] = s;
    __syncthreads();
    for (int o = 128; o > 0; o >>= 1) {
        if (threadIdx.x < o) sm[threadIdx.x] += sm[threadIdx.x + o];
        __syncthreads();
    }
    if (threadIdx.x == 0) *accOut = sm[0];
}

// ---------------- finalize: combine scales ---------------------------------
__global__ void lncc_finalize(const float* __restrict__ acc, float* __restrict__ out)
{
    const float w[3]  = {0.1f, 0.3f, 0.6f};
    const float nv[3] = {4.0f * 64 * 64 * 64, 4.0f * 28 * 28 * 28, 4.0f * 9 * 9 * 9};
    float t = 0.0f;
    for (int s = 0; s < 3; ++s) t += w[s] * (1.0f - acc[s] / nv[s]);
    out[0] = t * (float)NB;
}

// ---------------- per-scale launcher ---------------------------------------
template <int K, int ST, int O>
static void run_scale(const float* din, const float* dtg,
                      float* tmp1, float* tmp2, float* sums,
                      float* part, float* accSlot, hipStream_t stream)
{
    constexpr int TILES = (O + 15) / 16;
    constexpr int TOTW  = NLINES * O;
    constexpr int TOTD  = NB * O * O * O;
    constexpr int NBLK  = (TOTD + 255) / 256;       // <= 4096

#define LNCC_DO_FIELD(F)                                                        \
    lncc_pass_w<F, K, ST, O><<<(TOTW + 255) / 256, 256, 0, stream>>>(           \
        din, dtg, tmp1);                                                        \
    lncc_pass_h_wmma<K, ST, O><<<dim3(NSLABS, TILES * TILES), 32, 0, stream>>>( \
        tmp1, tmp2);                                                            \
    lncc_pass_d<K, ST, O><<<NBLK, 256, 0, stream>>>(                            \
        tmp2, sums + (size_t)(F) * SUMS_STRIDE);

    LNCC_DO_FIELD(0)
    LNCC_DO_FIELD(1)
    LNCC_DO_FIELD(2)
    LNCC_DO_FIELD(3)
    LNCC_DO_FIELD(4)
#undef LNCC_DO_FIELD

    lncc_partials<K, O><<<NBLK, 256, 0, stream>>>(sums, part);
    lncc_reduce<<<1, 256, 0, stream>>>(part, NBLK, accSlot);
}

// ---------------------------------------------------------------------------
extern "C" void kernel_launch(void* const* d_in, const int* in_sizes, int n_in,
                              void* d_out, int out_size, void* d_ws, size_t ws_size,
                              hipStream_t stream)
{
    (void)in_sizes; (void)n_in; (void)out_size; (void)ws_size;
    const float* din = (const float*)d_in[0];
    const float* dtg = (const float*)d_in[1];
    float* out = (float*)d_out;

    // workspace layout (floats), total ~51.7 MB
    const size_t TMP1 = (size_t)NB * NDIM * NDIM * 64;   // 5,308,416
    const size_t TMP2 = (size_t)NB * NDIM * 64 * 64;     // 2,359,296
    float* tmp1 = (float*)d_ws;
    float* tmp2 = tmp1 + TMP1;
    float* sums = tmp2 + TMP2;
    float* part = sums + 5 * (size_t)SUMS_STRIDE;
    float* acc  = part + 4096;

    run_scale< 9, 2, 64>(din, dtg, tmp1, tmp2, sums, part, acc + 0, stream);
    run_scale<18, 4, 28>(din, dtg, tmp1, tmp2, sums, part, acc + 1, stream);
    run_scale<36, 9,  9>(din, dtg, tmp1, tmp2, sums, part, acc + 2, stream);

    lncc_finalize<<<1, 1, 0, stream>>>(acc, out);
}